// CustomGatedGCNLayer_89129161327105
// MI455X (gfx1250) — compile-verified
//
#include <hip/hip_runtime.h>

typedef __attribute__((ext_vector_type(2))) float v2f;
typedef __attribute__((ext_vector_type(8))) float v8f;

#define NN 50000
#define EE 800000
#define DD 128
#define BN_EPS 1e-5f

// ---------------------------------------------------------------------------
// Zero scratch (hsum accumulator + BN stat counters, laid out contiguously)
// ---------------------------------------------------------------------------
__global__ void k_zero(float4* __restrict__ p, long n4) {
  long i = (long)blockIdx.x * blockDim.x + threadIdx.x;
  if (i < n4) p[i] = make_float4(0.f, 0.f, 0.f, 0.f);
}

// ---------------------------------------------------------------------------
// O[M x 128] = X[M x 128] @ W[128 x 128]^T + bias     (M multiple of 16)
// One 16-row tile per block; 8 waves, one 16-col tile each.
// fp32 WMMA 16x16x4 (memory-bound layer -> keep full precision on matrix pipe)
// ---------------------------------------------------------------------------
__global__ void __launch_bounds__(256) k_gemm16(
    const float* __restrict__ X, const float* __restrict__ W,
    const float* __restrict__ bias, float* __restrict__ O) {
  __shared__ __align__(16) float Xs[16 * DD];
  const int m0 = blockIdx.x << 4;
  {
    const float4* srcv = (const float4*)(X + (long)m0 * DD);
    float4* dstv = (float4*)Xs;
    dstv[threadIdx.x] = srcv[threadIdx.x];
    dstv[threadIdx.x + 256] = srcv[threadIdx.x + 256];
  }
  __syncthreads();

  const int wid  = threadIdx.x >> 5;
  const int lane = threadIdx.x & 31;
  const int half = lane >> 4;       // lane half selects K pair / M offset
  const int lid  = lane & 15;
  const int n    = (wid << 4) + lid;        // output column owned by this lane
  const float* wrow = W + (long)n * DD;     // row n of W (O = X @ W^T)
  __builtin_prefetch(wrow, 0, 0);           // global_prefetch_b8

  v8f acc = {};
#pragma unroll
  for (int k = 0; k < DD; k += 4) {
    // A frag: lanes 0-15 hold K={k,k+1}, lanes 16-31 hold K={k+2,k+3} of row m0+lid
    v2f a = *(const v2f*)&Xs[lid * DD + k + 2 * half];
    // B frag: same pairing on W row n (B = W^T, K x 16)
    v2f b = *(const v2f*)&wrow[k + 2 * half];
    acc = __builtin_amdgcn_wmma_f32_16x16x4_f32(
        /*neg_a=*/false, a, /*neg_b=*/false, b,
        /*c_mod=*/(short)0, acc, /*reuse_a=*/false, /*reuse_b=*/false);
  }

  const float bv = bias[n];
#pragma unroll
  for (int v = 0; v < 8; ++v) {
    const int m = m0 + v + (half << 3);     // C/D layout: M = v + 8*half
    O[(long)m * DD + n] = acc[v] + bv;
  }
}

// ---------------------------------------------------------------------------
// Edge stage: DEh, he, e_new (in place over Ce), sigma/gate, scatter-add,
// BN column stats for e (register-accumulated, one atomic per thread).
// 128 threads per edge, 2 edges per 256-thread block per iteration.
// ---------------------------------------------------------------------------
__global__ void __launch_bounds__(256) k_edge(
    const float* __restrict__ Bh, const float* __restrict__ Dh,
    const float* __restrict__ Ehm, float* __restrict__ CeEn,
    const int* __restrict__ src, const int* __restrict__ dst,
    const int* __restrict__ smask, const float* __restrict__ diff,
    const float* __restrict__ dist, const float* __restrict__ dirn,
    float* __restrict__ hsum, float* __restrict__ stats) {
  __shared__ float red[8];
  const int t   = threadIdx.x & (DD - 1);   // feature
  const int sub = threadIdx.x >> 7;         // which edge of the pair
  const int wv  = threadIdx.x >> 5;
  const int ln  = threadIdx.x & 31;
  float lsum = 0.f, lsq = 0.f;

  for (long p = blockIdx.x; p < (long)EE / 2; p += gridDim.x) {
    const long eid = (p << 1) + sub;
    const int s = src[eid], d = dst[eid];
    const float de = Dh[(long)d * DD + t] - Ehm[(long)s * DD + t];
    const float he = Bh[(long)s * DD + t] + de;
    const float en = de + CeEn[eid * DD + t];

    // row norm of e_new over 128 features: wave32 shuffle + 4-partial LDS combine
    float sq = en * en;
#pragma unroll
    for (int off = 16; off; off >>= 1) sq += __shfl_down(sq, off, 32);
    if (ln == 0) red[wv] = sq;
    __syncthreads();
    const float sumsq = red[(sub << 2) + 0] + red[(sub << 2) + 1] +
                        red[(sub << 2) + 2] + red[(sub << 2) + 3];
    __syncthreads();

    const float sigma = __expf(-0.5f * sqrtf(sumsq));
    const float ds  = dist[eid];
    const float inv = ds > 0.f ? 1.f / ds : 1.f;
    const float ex = diff[eid * 2 + 0] * inv, ey = diff[eid * 2 + 1] * inv;
    const float vdx = dirn[(long)d * 2 + 0], vdy = dirn[(long)d * 2 + 1];
    const float uv  = dirn[(long)s * 2 + 0] * vdx + dirn[(long)s * 2 + 1] * vdy;
    const float ehd = ex * vdx + ey * vdy;
    const float f   = 0.5f * (1.f - ehd) * uv * sigma;
    const float gate = (smask[eid] == 1) ? f : sigma;

    atomicAdd(&hsum[(long)d * DD + t], he * gate);
    CeEn[eid * DD + t] = en;                // e_new overwrites Ce in place
    lsum += en; lsq += en * en;
  }
  atomicAdd(&stats[t], lsum);        // e column sum
  atomicAdd(&stats[DD + t], lsq);    // e column sumsq
}

// ---------------------------------------------------------------------------
// h_pre = Ah + hsum (in place over hsum) + BN column stats for h
// ---------------------------------------------------------------------------
__global__ void __launch_bounds__(256) k_nodepre(
    const float* __restrict__ Ah, float* __restrict__ hsum,
    float* __restrict__ stats) {
  const int t   = threadIdx.x & (DD - 1);
  const int sub = threadIdx.x >> 7;
  float ls = 0.f, lq = 0.f;
  for (long p = blockIdx.x; p < (long)NN / 2; p += gridDim.x) {
    const long n = (p << 1) + sub;
    const float v = Ah[n * DD + t] + hsum[n * DD + t];
    hsum[n * DD + t] = v;
    ls += v; lq += v * v;
  }
  atomicAdd(&stats[256 + t], ls);    // h column sum
  atomicAdd(&stats[384 + t], lq);    // h column sumsq
}

// ---------------------------------------------------------------------------
// mu / inv-std for both BatchNorms
// stats: [0)e_sum [128)e_sq [256)h_sum [384)h_sq [512)h_mu [640)h_istd
//        [768)e_mu [896)e_istd
// ---------------------------------------------------------------------------
__global__ void k_finalize(float* __restrict__ stats) {
  const int t = threadIdx.x;
  if (t < DD) {
    const float mu  = stats[256 + t] / (float)NN;
    const float var = stats[384 + t] / (float)NN - mu * mu;
    stats[512 + t] = mu;
    stats[640 + t] = rsqrtf(var + BN_EPS);
  } else {
    const int u = t - DD;
    const float mu  = stats[u] / (float)EE;
    const float var = stats[DD + u] / (float)EE - mu * mu;
    stats[768 + u] = mu;
    stats[896 + u] = rsqrtf(var + BN_EPS);
  }
}

// ---------------------------------------------------------------------------
// out = resid + relu(gamma*(pre - mu)*istd + beta), float4-vectorized
// ---------------------------------------------------------------------------
__global__ void __launch_bounds__(256) k_norm_out(
    const float* __restrict__ pre, const float* __restrict__ resid,
    const float* __restrict__ gamma, const float* __restrict__ beta,
    const float* __restrict__ mu, const float* __restrict__ istd,
    float* __restrict__ out, long n4) {
  const long i = (long)blockIdx.x * blockDim.x + threadIdx.x;
  if (i >= n4) return;
  const int t = (int)((i << 2) & (DD - 1));
  const float4 p = ((const float4*)pre)[i];
  const float4 r = ((const float4*)resid)[i];
  float4 o;
  o.x = r.x + fmaxf(0.f, gamma[t + 0] * (p.x - mu[t + 0]) * istd[t + 0] + beta[t + 0]);
  o.y = r.y + fmaxf(0.f, gamma[t + 1] * (p.y - mu[t + 1]) * istd[t + 1] + beta[t + 1]);
  o.z = r.z + fmaxf(0.f, gamma[t + 2] * (p.z - mu[t + 2]) * istd[t + 2] + beta[t + 2]);
  o.w = r.w + fmaxf(0.f, gamma[t + 3] * (p.w - mu[t + 3]) * istd[t + 3] + beta[t + 3]);
  ((float4*)out)[i] = o;
}

// ---------------------------------------------------------------------------
extern "C" void kernel_launch(void* const* d_in, const int* in_sizes, int n_in,
                              void* d_out, int out_size, void* d_ws, size_t ws_size,
                              hipStream_t stream) {
  const float* h     = (const float*)d_in[0];
  const float* e     = (const float*)d_in[1];
  const int*   src   = (const int*)d_in[2];
  const int*   dst   = (const int*)d_in[3];
  const int*   smask = (const int*)d_in[4];
  const float* diff  = (const float*)d_in[5];
  const float* dist  = (const float*)d_in[6];
  const float* dirn  = (const float*)d_in[7];
  const float* WA = (const float*)d_in[8];  const float* bA = (const float*)d_in[9];
  const float* WB = (const float*)d_in[10]; const float* bB = (const float*)d_in[11];
  const float* WC = (const float*)d_in[12]; const float* bC = (const float*)d_in[13];
  const float* WD = (const float*)d_in[14]; const float* bD = (const float*)d_in[15];
  const float* WE = (const float*)d_in[16]; const float* bE = (const float*)d_in[17];
  const float* gamma_h = (const float*)d_in[18]; const float* beta_h = (const float*)d_in[19];
  const float* gamma_e = (const float*)d_in[20]; const float* beta_e = (const float*)d_in[21];

  float* ws    = (float*)d_ws;
  float* Ah    = ws;
  float* Bh    = ws + 1L * NN * DD;
  float* Dh    = ws + 2L * NN * DD;
  float* Eh    = ws + 3L * NN * DD;
  float* hsum  = ws + 4L * NN * DD;   // becomes h_pre in place
  float* stats = ws + 5L * NN * DD;   // 1024 floats
  float* Ce    = stats + 1024;        // E*D, becomes e_new in place

  float* hout = (float*)d_out;
  float* eout = hout + (long)NN * DD;

  // 1) zero accumulators (hsum + stats are contiguous)
  const long z4 = ((long)NN * DD + 1024) / 4;
  k_zero<<<(unsigned)((z4 + 255) / 256), 256, 0, stream>>>((float4*)hsum, z4);

  // 2) WMMA GEMMs
  k_gemm16<<<NN / 16, 256, 0, stream>>>(h, WA, bA, Ah);
  k_gemm16<<<NN / 16, 256, 0, stream>>>(h, WB, bB, Bh);
  k_gemm16<<<NN / 16, 256, 0, stream>>>(h, WD, bD, Dh);
  k_gemm16<<<NN / 16, 256, 0, stream>>>(h, WE, bE, Eh);
  k_gemm16<<<EE / 16, 256, 0, stream>>>(e, WC, bC, Ce);

  // 3) edge stage
  k_edge<<<4096, 256, 0, stream>>>(Bh, Dh, Eh, Ce, src, dst, smask,
                                   diff, dist, dirn, hsum, stats);
  // 4) node pre-activation + stats
  k_nodepre<<<2048, 256, 0, stream>>>(Ah, hsum, stats);
  // 5) BN stat finalize
  k_finalize<<<1, 256, 0, stream>>>(stats);
  // 6) normalize + relu + residual
  const long h4 = (long)NN * DD / 4;
  k_norm_out<<<(unsigned)((h4 + 255) / 256), 256, 0, stream>>>(
      hsum, h, gamma_h, beta_h, stats + 512, stats + 640, hout, h4);
  const long e4 = (long)EE * DD / 4;
  k_norm_out<<<(unsigned)((e4 + 255) / 256), 256, 0, stream>>>(
      Ce, e, gamma_e, beta_e, stats + 768, stats + 896, eout, e4);
}